// SimilarityLayer_68676527063498
// MI455X (gfx1250) — compile-verified
//
#include <hip/hip_runtime.h>

typedef float v2f __attribute__((ext_vector_type(2)));
typedef float v4f __attribute__((ext_vector_type(4)));
typedef float v8f __attribute__((ext_vector_type(8)));

static constexpr int Bn = 64;
static constexpr int Tn = 2048;
static constexpr int Jn = 128;
static constexpr int Dn = 256;

static constexpr int KC    = 32;   // K-chunk staged in LDS per iteration
static constexpr int LDSTR = 36;   // 32 + 4 floats pad -> conflict-free b64 frag reads
static constexpr int TROWS = 256;  // T rows per block (8 waves x 2 tiles x 16)

__global__ __launch_bounds__(256)
void SimilarityLayer_wmma_f32_kernel(const float* __restrict__ H,
                                     const float* __restrict__ U,
                                     const float* __restrict__ w,
                                     float* __restrict__ out) {
    __shared__ float Hs[TROWS * LDSTR];   // (H * w3) chunk, padded      (36 KB)
    __shared__ float Us[Jn * LDSTR];      // raw U chunk, padded         (18 KB)
    __shared__ float sh[TROWS];           // H . w1 per local t-row
    __shared__ float su[Jn];              // U . w2 per j

    const int tid  = threadIdx.x;
    const int wave = tid >> 5;
    const int lane = tid & 31;
    const int b    = blockIdx.y;
    const int t0g  = blockIdx.x * TROWS;  // first global t-row of this block

    const float* __restrict__ w1 = w;
    const float* __restrict__ w2 = w + Dn;
    const float* __restrict__ w3 = w + 2 * Dn;

    const float* __restrict__ Hb = H + (size_t)b * Tn * Dn + (size_t)t0g * Dn;
    const float* __restrict__ Ub = U + (size_t)b * Jn * Dn;

    // zero the bias accumulators
    sh[tid] = 0.0f;
    if (tid < Jn) su[tid] = 0.0f;
    __syncthreads();

    // 2 M-tiles x 8 J-tiles of 16x16 f32 accumulators (128 VGPRs)
    v8f acc0[8] = {};
    v8f acc1[8] = {};

    const int n     = lane & 15;          // N column within tile / M row for A
    const int khalf = (lane >> 4) << 1;   // 0 or 2 : K-pair selected by lane half

    const float* __restrict__ hrow0 = Hs + (wave * 32 + n) * LDSTR;
    const float* __restrict__ hrow1 = hrow0 + 16 * LDSTR;

    const int rbase = tid >> 3;           // staging row base (0..31)
    const int kq    = (tid & 7) << 2;     // staging k offset 0,4,...,28 (chunk-local)

    #pragma unroll 1
    for (int kc = 0; kc < Dn; kc += KC) {
        // ---- stage H*w3 (256x32) and U (128x32) into LDS, fold bias dots ----
        const v4f w1v = *(const v4f*)(w1 + kc + kq);
        const v4f w2v = *(const v4f*)(w2 + kc + kq);
        const v4f w3v = *(const v4f*)(w3 + kc + kq);

        #pragma unroll
        for (int i = 0; i < 8; ++i) {
            const int r = rbase + i * 32;                      // 0..255
            const v4f h4 = *(const v4f*)(Hb + (size_t)r * Dn + kc + kq);
            const float dh = h4.x * w1v.x + h4.y * w1v.y + h4.z * w1v.z + h4.w * w1v.w;
            atomicAdd(&sh[r], dh);                             // ds_add_f32
            *(v4f*)(Hs + r * LDSTR + kq) = h4 * w3v;           // premultiplied A panel
        }
        #pragma unroll
        for (int i = 0; i < 4; ++i) {
            const int r = rbase + i * 32;                      // 0..127
            const v4f u4 = *(const v4f*)(Ub + (size_t)r * Dn + kc + kq);
            const float du = u4.x * w2v.x + u4.y * w2v.y + u4.z * w2v.z + u4.w * w2v.w;
            atomicAdd(&su[r], du);                             // ds_add_f32
            *(v4f*)(Us + r * LDSTR + kq) = u4;                 // raw B panel
        }
        __syncthreads();

        // ---- WMMA core: 8 K-steps x (2 M-tiles x 8 J-tiles) per chunk ----
        #pragma unroll
        for (int kk = 0; kk < KC; kk += 4) {
            const v2f a0 = *(const v2f*)(hrow0 + kk + khalf);  // A frag, tile 0
            const v2f a1 = *(const v2f*)(hrow1 + kk + khalf);  // A frag, tile 1
            #pragma unroll
            for (int jt = 0; jt < 8; ++jt) {
                const v2f bf = *(const v2f*)(Us + (jt * 16 + n) * LDSTR + kk + khalf);
                acc0[jt] = __builtin_amdgcn_wmma_f32_16x16x4_f32(
                    false, a0, false, bf, (short)0, acc0[jt], false, false);
                acc1[jt] = __builtin_amdgcn_wmma_f32_16x16x4_f32(
                    false, a1, false, bf, (short)0, acc1[jt], false, false);
            }
        }
        __syncthreads();
    }

    // ---- epilogue: add rank-1 biases and store ----
    const int rh = (lane >> 4) << 3;       // 0 or 8 (M offset of lane half)

    #pragma unroll
    for (int sub = 0; sub < 2; ++sub) {
        const int trowbase = wave * 32 + sub * 16 + rh;   // local t-row of acc r=0
        const v4f s0 = *(const v4f*)(sh + trowbase);
        const v4f s1 = *(const v4f*)(sh + trowbase + 4);
        const float shv[8] = {s0.x, s0.y, s0.z, s0.w, s1.x, s1.y, s1.z, s1.w};

        float* __restrict__ outBase =
            out + (size_t)b * Tn * Jn + (size_t)(t0g + trowbase) * Jn;
        const v8f* accp = sub ? acc1 : acc0;

        #pragma unroll
        for (int jt = 0; jt < 8; ++jt) {
            const float suv = su[jt * 16 + n];
            const v8f a = accp[jt];
            #pragma unroll
            for (int r = 0; r < 8; ++r) {
                outBase[(size_t)r * Jn + jt * 16 + n] = a[r] + shv[r] + suv;
            }
        }
    }
}

extern "C" void kernel_launch(void* const* d_in, const int* in_sizes, int n_in,
                              void* d_out, int out_size, void* d_ws, size_t ws_size,
                              hipStream_t stream) {
    (void)in_sizes; (void)n_in; (void)out_size; (void)d_ws; (void)ws_size;
    const float* H = (const float*)d_in[0];
    const float* U = (const float*)d_in[1];
    const float* w = (const float*)d_in[2];
    float* out     = (float*)d_out;

    dim3 grid(Tn / TROWS, Bn);   // (8, 64)
    dim3 block(256);             // 8 waves (wave32)
    SimilarityLayer_wmma_f32_kernel<<<grid, block, 0, stream>>>(H, U, w, out);
}